// NeuroRVQTokenizer_4982162063517
// MI455X (gfx1250) — compile-verified
//
#include <hip/hip_runtime.h>
#include <hip/hip_bf16.h>

typedef __attribute__((ext_vector_type(2))) float v2f;
typedef __attribute__((ext_vector_type(8))) float v8f;

#define NQ 8
#define NV 8192
#define DD 200

// ---------------- conv1: Cin=1, per-branch ----------------
__global__ __launch_bounds__(256) void conv1_kernel(const float* __restrict__ x,
                                                    const float* __restrict__ w,
                                                    const float* __restrict__ bias,
                                                    float* __restrict__ y, int K) {
    int idx = blockIdx.x * 256 + threadIdx.x;
    const int total = 8 * 8 * 128 * 200;
    if (idx >= total) return;
    int t  = idx % 200;
    int na = (idx / 200) & 127;
    int c  = (idx / (200 * 128)) & 7;
    int b  = idx / (200 * 128 * 8);
    int pad = (K - 1) >> 1;
    float s = bias[c];
    const float* xr = x + ((size_t)b * 128 + na) * 200;
    for (int k = 0; k < K; ++k) {
        int tt = t + k - pad;
        if (tt >= 0 && tt < 200) s += xr[tt] * w[c * K + k];
    }
    y[idx] = s;
}

// ---------------- group-norm stats: per (br,b,g) over 2*128*T ----------------
__global__ __launch_bounds__(256) void gn_stats_kernel(const float* __restrict__ y,
                                                       float* __restrict__ stats, int T) {
    int blk = blockIdx.x;          // br*32 + b*4 + g
    int g  = blk & 3;
    int b  = (blk >> 2) & 7;
    int br = blk >> 5;
    const float* base = y + ((((size_t)br * 8 + b) * 8) + g * 2) * 128 * T;
    int n = 2 * 128 * T;
    float s = 0.f, s2 = 0.f;
    for (int i = threadIdx.x; i < n; i += 256) {
        float v = base[i];
        s += v; s2 += v * v;
    }
    __shared__ float sh[256], sh2[256];
    sh[threadIdx.x] = s; sh2[threadIdx.x] = s2;
    __syncthreads();
    for (int off = 128; off > 0; off >>= 1) {
        if (threadIdx.x < off) {
            sh[threadIdx.x]  += sh[threadIdx.x + off];
            sh2[threadIdx.x] += sh2[threadIdx.x + off];
        }
        __syncthreads();
    }
    if (threadIdx.x == 0) {
        float m = sh[0] / n;
        float var = sh2[0] / n - m * m;
        stats[blk * 2] = m;
        stats[blk * 2 + 1] = var;
    }
}

__device__ __forceinline__ float gelu_exact(float v) {
    return 0.5f * v * (1.0f + erff(v * 0.70710678118654752f));
}

// ---------------- GN + GELU + avg-pool(pk): [4][8][8][128][T] -> [...][T/pk] ----------------
__global__ __launch_bounds__(256) void gn_gelu_pool_kernel(const float* __restrict__ y,
                                                           const float* __restrict__ stats,
                                                           const float* __restrict__ gamma,
                                                           const float* __restrict__ beta,
                                                           float* __restrict__ out,
                                                           int T, int pk) {
    int To = T / pk;
    int total = 4 * 8 * 8 * 128 * To;
    int idx = blockIdx.x * 256 + threadIdx.x;
    if (idx >= total) return;
    int t  = idx % To;
    int na = (idx / To) & 127;
    int c  = (idx / (To * 128)) & 7;
    int b  = (idx / (To * 128 * 8)) & 7;
    int br = idx / (To * 128 * 8 * 8);
    int g  = c >> 1;
    int sb = ((br * 8 + b) * 4 + g) * 2;
    float m = stats[sb], var = stats[sb + 1];
    float inv = rsqrtf(var + 1e-5f);
    float ga = gamma[br * 8 + c], be = beta[br * 8 + c];
    const float* yr = y + ((((size_t)br * 8 + b) * 8 + c) * 128 + na) * T + t * pk;
    float acc = 0.f;
    for (int j = 0; j < pk; ++j) {
        float v = (yr[j] - m) * inv * ga + be;
        acc += gelu_exact(v);
    }
    out[idx] = acc / pk;
}

// ---------------- conv2: Cin=8, per-branch, T=100 ----------------
__global__ __launch_bounds__(256) void conv2_kernel(const float* __restrict__ y,
                                                    const float* __restrict__ w,
                                                    const float* __restrict__ bias,
                                                    float* __restrict__ out, int K) {
    int idx = blockIdx.x * 256 + threadIdx.x;
    const int total = 8 * 8 * 128 * 100;
    if (idx >= total) return;
    int t  = idx % 100;
    int na = (idx / 100) & 127;
    int co = (idx / (100 * 128)) & 7;
    int b  = idx / (100 * 128 * 8);
    int pad = (K - 1) >> 1;
    float s = bias[co];
    for (int ci = 0; ci < 8; ++ci) {
        const float* yr = y + (((size_t)b * 8 + ci) * 128 + na) * 100;
        const float* wr = w + (co * 8 + ci) * K;
        for (int k = 0; k < K; ++k) {
            int tt = t + k - pad;
            if (tt >= 0 && tt < 100) s += yr[tt] * wr[k];
        }
    }
    out[idx] = s;
}

// ---------------- GN + GELU + pool4 + transpose -> feats [4][8][128][200] ----------------
__global__ __launch_bounds__(256) void gn_gelu_pool_tr_kernel(const float* __restrict__ y,
                                                              const float* __restrict__ stats,
                                                              const float* __restrict__ gamma,
                                                              const float* __restrict__ beta,
                                                              float* __restrict__ feats) {
    int total = 4 * 8 * 8 * 128 * 25;
    int idx = blockIdx.x * 256 + threadIdx.x;
    if (idx >= total) return;
    int t  = idx % 25;
    int na = (idx / 25) & 127;
    int c  = (idx / (25 * 128)) & 7;
    int b  = (idx / (25 * 128 * 8)) & 7;
    int br = idx / (25 * 128 * 8 * 8);
    int g  = c >> 1;
    int sb = ((br * 8 + b) * 4 + g) * 2;
    float m = stats[sb], var = stats[sb + 1];
    float inv = rsqrtf(var + 1e-5f);
    float ga = gamma[br * 8 + c], be = beta[br * 8 + c];
    const float* yr = y + ((((size_t)br * 8 + b) * 8 + c) * 128 + na) * 100 + t * 4;
    float acc = 0.f;
    for (int j = 0; j < 4; ++j) {
        float v = (yr[j] - m) * inv * ga + be;
        acc += gelu_exact(v);
    }
    feats[(((size_t)br * 8 + b) * 128 + na) * DD + t * 8 + c] = acc * 0.25f;
}

// ---------------- 0.5*||c||^2 per code ----------------
__global__ __launch_bounds__(256) void cnorm_kernel(const float* __restrict__ cbs,
                                                    float* __restrict__ cn) {
    int v = blockIdx.x * 256 + threadIdx.x;   // 0 .. 4*NQ*NV-1
    if (v >= 4 * NQ * NV) return;
    const float* c = cbs + (size_t)v * DD;
    float s = 0.f;
    for (int k = 0; k < DD; ++k) s += c[k] * c[k];
    cn[v] = 0.5f * s;
}

// ---------------- RVQ: WMMA f32 16x16x4 distance GEMM + argmin + residual ----------------
// grid: 256 workgroups = 4 branches * 64 M-tiles of 16 rows; 256 threads = 8 waves
__global__ __launch_bounds__(256) void rvq_kernel(const float* __restrict__ feats,
                                                  const float* __restrict__ cbs,
                                                  const float* __restrict__ cnorms,
                                                  float* __restrict__ out,
                                                  int* __restrict__ idxout) {
    int wg = blockIdx.x;
    int br = wg >> 6;
    int mt = wg & 63;
    int tid = threadIdx.x;
    int lane = tid & 31;
    int wave = tid >> 5;
    int lhalf = lane >> 4;     // 0: lanes 0-15, 1: lanes 16-31
    int lmod  = lane & 15;

    __shared__ float R[16 * DD];         // residual rows
    __shared__ float redv[16][128];      // argmin candidates (value)
    __shared__ int   redi[16][128];      // argmin candidates (index)
    __shared__ int   chosen[16];

    const float* fbase = feats + ((size_t)br * 1024 + mt * 16) * DD;
    for (int i = tid; i < 16 * DD; i += 256) R[i] = fbase[i];
    __syncthreads();

    const float* cb_br = cbs + (size_t)br * NQ * NV * DD;
    const float* cn_br = cnorms + (size_t)br * NQ * NV;

    int row = tid >> 4;        // for the two-level reduction
    int slot = tid & 15;

    for (int q = 0; q < NQ; ++q) {
        const float* cb = cb_br + (size_t)q * NV * DD;
        const float* cn = cn_br + (size_t)q * NV;

        float minv[8];
        int   mini[8];
#pragma unroll
        for (int v = 0; v < 8; ++v) { minv[v] = 3.4e38f; mini[v] = 0; }

        // A fragment source: R row (lane%16), k = 4s + (lane/16)*2 .. +1  (float2, 8B aligned)
        const float* arow = &R[lmod * DD + lhalf * 2];

        for (int nt = wave; nt < NV / 16; nt += 8) {
            int nbase = nt * 16;
            // B fragment source: code row (nbase + lane%16), same k pattern (B^T laid out like A)
            const float* crow = cb + (size_t)(nbase + lmod) * DD + lhalf * 2;

            v8f acc = {0.f, 0.f, 0.f, 0.f, 0.f, 0.f, 0.f, 0.f};
            for (int s = 0; s < 50; ++s) {
                v2f a; a.x = arow[4 * s]; a.y = arow[4 * s + 1];
                v2f b; b.x = crow[4 * s]; b.y = crow[4 * s + 1];
                acc = __builtin_amdgcn_wmma_f32_16x16x4_f32(
                    false, a, false, b, (short)0, acc, false, false);
            }
            int n = nbase + lmod;
            float c2 = cn[n];
            // Branchless running argmin. Per lane n is strictly increasing across
            // nt, so strict '<' keeps the first (lowest-index) minimum -> matches
            // jnp.argmin tie-breaking within a lane. EXEC stays all-ones.
#pragma unroll
            for (int v = 0; v < 8; ++v) {
                float sc = c2 - acc[v];          // acc[v]=dot(R[m],C[n]), m=v+8*lhalf
                bool lt = sc < minv[v];
                mini[v] = lt ? n : mini[v];
                minv[v] = lt ? sc : minv[v];
            }
        }
        // deposit candidates: row m = v + 8*lhalf, slot = wave*16 + lmod
#pragma unroll
        for (int v = 0; v < 8; ++v) {
            int m = v + 8 * lhalf;
            redv[m][wave * 16 + lmod] = minv[v];
            redi[m][wave * 16 + lmod] = mini[v];
        }
        __syncthreads();
        // level 1: 256 threads, thread (row,slot) reduces slots slot, slot+16, ... slot+112
        {
            float bv = redv[row][slot]; int bi = redi[row][slot];
#pragma unroll
            for (int j = 16; j < 128; j += 16) {
                float vv = redv[row][slot + j]; int ii = redi[row][slot + j];
                bool better = (vv < bv) || (vv == bv && ii < bi);
                bv = better ? vv : bv;
                bi = better ? ii : bi;
            }
            redv[row][slot] = bv;
            redi[row][slot] = bi;
        }
        __syncthreads();
        // level 2: 16 threads finish 16 slots each
        if (tid < 16) {
            float bv = redv[tid][0]; int bi = redi[tid][0];
#pragma unroll
            for (int j = 1; j < 16; ++j) {
                float vv = redv[tid][j]; int ii = redi[tid][j];
                bool better = (vv < bv) || (vv == bv && ii < bi);
                bv = better ? vv : bv;
                bi = better ? ii : bi;
            }
            chosen[tid] = bi;
            int rowg = mt * 16 + tid;
            int b = rowg >> 7, na = rowg & 127;
            idxout[((br * NQ + q) * 8 + b) * 128 + na] = bi;
        }
        __syncthreads();
        // residual update: R[m] -= C[chosen[m]]  (no integer division)
        if (tid < DD) {
#pragma unroll
            for (int m = 0; m < 16; ++m) {
                R[m * DD + tid] -= cb[(size_t)chosen[m] * DD + tid];
            }
        }
        __syncthreads();
    }

    // out = feat - final residual = sum of selected codes
    float* obase = out + ((size_t)br * 1024 + mt * 16) * DD;
    for (int i = tid; i < 16 * DD; i += 256) obase[i] = fbase[i] - R[i];
}

extern "C" void kernel_launch(void* const* d_in, const int* in_sizes, int n_in,
                              void* d_out, int out_size, void* d_ws, size_t ws_size,
                              hipStream_t stream) {
    const float* x    = (const float*)d_in[0];
    const float* w1[4] = {(const float*)d_in[1], (const float*)d_in[2],
                          (const float*)d_in[3], (const float*)d_in[4]};
    const float* b1  = (const float*)d_in[5];
    const float* g1  = (const float*)d_in[6];
    const float* be1 = (const float*)d_in[7];
    const float* w2[4] = {(const float*)d_in[8], (const float*)d_in[9],
                          (const float*)d_in[10], (const float*)d_in[11]};
    const float* b2  = (const float*)d_in[12];
    const float* g2  = (const float*)d_in[13];
    const float* be2 = (const float*)d_in[14];
    const float* cbs = (const float*)d_in[15];

    // workspace carve-up (floats)
    float* y1     = (float*)d_ws;                 // 4*8*8*128*200 = 6,553,600
    float* stats1 = y1 + 6553600;                 // 256
    float* y2     = stats1 + 256;                 // 4*8*8*128*100 = 3,276,800
    float* y3     = y2 + 3276800;                 // 3,276,800
    float* stats2 = y3 + 3276800;                 // 256
    float* feats  = stats2 + 256;                 // 4*8*128*200 = 819,200
    float* cnorms = feats + 819200;               // 4*8*8192 = 262,144

    const int K1[4] = {21, 15, 9, 5};
    const int K2[4] = {9, 7, 5, 3};

    // stage 1 conv (per branch)
    for (int br = 0; br < 4; ++br) {
        conv1_kernel<<<6400, 256, 0, stream>>>(x, w1[br], b1 + br * 8,
                                               y1 + (size_t)br * 1638400, K1[br]);
    }
    gn_stats_kernel<<<128, 256, 0, stream>>>(y1, stats1, 200);
    gn_gelu_pool_kernel<<<12800, 256, 0, stream>>>(y1, stats1, g1, be1, y2, 200, 2);

    // stage 2 conv (per branch)
    for (int br = 0; br < 4; ++br) {
        conv2_kernel<<<3200, 256, 0, stream>>>(y2 + (size_t)br * 819200, w2[br],
                                               b2 + br * 8, y3 + (size_t)br * 819200, K2[br]);
    }
    gn_stats_kernel<<<128, 256, 0, stream>>>(y3, stats2, 100);
    gn_gelu_pool_tr_kernel<<<3200, 256, 0, stream>>>(y3, stats2, g2, be2, feats);

    // code norms
    cnorm_kernel<<<1024, 256, 0, stream>>>(cbs, cnorms);

    // RVQ with WMMA
    float* out_f = (float*)d_out;                         // [4][8][128][200]
    int* idx_out = (int*)(out_f + 4 * 8 * 128 * 200);     // [4][8][8][128]
    rvq_kernel<<<256, 256, 0, stream>>>(feats, cbs, cnorms, out_f, idx_out);
}